// MixQuantize_21620865368348
// MI455X (gfx1250) — compile-verified
//
#include <hip/hip_runtime.h>
#include <hip/hip_bf16.h>

// Sizes from the reference
#define B_    16
#define NH    256
#define NE    8192
#define ED    256
#define HWD   16
#define L     256      // HW*HW pixels per batch
#define LEN_KEEP 128
#define KLW   0.0005f

typedef float v2f __attribute__((ext_vector_type(2)));
typedef float v4f __attribute__((ext_vector_type(4)));
typedef float v8f __attribute__((ext_vector_type(8)));

// -------- workspace layout (floats) --------
// [0, 4096)      mask (0/1 float) per pixel
// [4096, 8192)   ind  (int32)     per pixel
// [8192]         loss accumulator (float, atomicAdd target)

// ---------------------------------------------------------------------------
// Kernel 1: per-batch rank mask (stable argsort semantics) + zero loss accum.
// ---------------------------------------------------------------------------
__global__ __launch_bounds__(256) void mask_kernel(const float* __restrict__ noise,
                                                   float* __restrict__ mask,
                                                   float* __restrict__ accum) {
  __shared__ float s[L];
  const int b = blockIdx.x, j = threadIdx.x;
  s[j] = noise[b * L + j];
  __syncthreads();
  const float v = s[j];
  int rank = 0;
  for (int i = 0; i < L; ++i) {
    float o = s[i];
    rank += (o < v) || (o == v && i < j);   // stable-sort tiebreak
  }
  mask[b * L + j] = (rank >= LEN_KEEP) ? 1.0f : 0.0f;
  if (b == 0 && j == 0) *accum = 0.0f;
}

// ---------------------------------------------------------------------------
// Kernel 2: fused  logits-GEMM (WMMA f32 16x16x4)  +  gumbel  +  dual argmax
//           +  softmax KL accumulation.  One workgroup = 16 pixels (one batch,
//           one 16-wide hw strip); 8 waves sweep NE=8192 in interleaved
//           16-column tiles. No logits tensor is ever written to memory.
// ---------------------------------------------------------------------------
__global__ __launch_bounds__(256) void quant_kernel(
    const float* __restrict__ z, const float* __restrict__ proj_w,
    const float* __restrict__ proj_b, const float* __restrict__ gumbel,
    const float* __restrict__ mask, int* __restrict__ ind_ws,
    int* __restrict__ ind_out, float* __restrict__ accum) {
  // A tile (16 pixels x 256 channels), pitch 260 floats -> conflict-free
  // ds_load_2addr_b64 WMMA-A fragment reads (bank = 4*m{+0,+1} vs 4*m{+2,+3}).
  __shared__ float A[HWD * 260];
  __shared__ float fstat[8][16][4];  // per wave, per row: stMax, dtMax, S, T
  __shared__ int   istat[8][16][2];  // per wave, per row: stIdx, dtIdx

  const int g   = blockIdx.x;        // 0..255
  const int b   = g >> 4;
  const int hw0 = (g & 15) << 4;
  const int t   = threadIdx.x;
  const int wv  = t >> 5;            // wave 0..7
  const int l   = t & 31;            // lane
  const int hi  = l >> 4;            // half (0: rows 0-7 / K even-pair, 1: rows 8-15)
  const int ln  = l & 15;

  // Stage A = z[b, :, hw0:hw0+16] transposed to [m][c].
  {
    const float* zp = z + ((size_t)b * NH) * L + (size_t)t * L + hw0;
    #pragma unroll
    for (int m = 0; m < 16; ++m) A[m * 260 + t] = zp[m];
  }
  __syncthreads();

  // Per-lane running stats for this lane's 8 rows (m = 8*hi + j) over the
  // lane's private column stream (n = 16*ti + ln, strictly increasing -> '>'
  // keeps the first max, matching jnp.argmax within the stream).
  float stMax[8], dtMax[8], S[8], T[8];
  int stIdx[8], dtIdx[8];
  #pragma unroll
  for (int j = 0; j < 8; ++j) {
    stMax[j] = -3.4e38f; dtMax[j] = -3.4e38f;
    S[j] = 0.0f; T[j] = 0.0f; stIdx[j] = 0x7fffffff; dtIdx[j] = 0x7fffffff;
  }

  const float* aBase = &A[ln * 260 + 2 * hi];

  for (int ti = wv; ti < NE / 16; ti += 8) {
    const int n0 = ti * 16;
    const int n  = n0 + ln;             // this lane's column

    // --- Issue the tile's HBM traffic FIRST so the whole 64-WMMA chain hides
    // its latency: gumbel (streamed NT, 134 MB single-use) + proj_b.
    const float* gp = gumbel + (((size_t)b * NE + n) * L) + hw0 + 8 * hi;
    v4f g0 = __builtin_nontemporal_load((const v4f*)gp);
    v4f g1 = __builtin_nontemporal_load(((const v4f*)gp) + 1);
    const float pb = proj_b[n];

    v8f c = {0.f, 0.f, 0.f, 0.f, 0.f, 0.f, 0.f, 0.f};
    const float* bptr = proj_w + (size_t)n * NH + 2 * hi;  // B[k][n] = proj_w[n][k]
    #pragma unroll 16
    for (int kb = 0; kb < NH; kb += 4) {
      v2f a  = *(const v2f*)(aBase + kb);   // ds_load_2addr_b64, conflict-free
      v2f bb = *(const v2f*)(bptr + kb);    // L2-resident proj_w
      // 8 args: (neg_a, A, neg_b, B, c_mod, C, reuse_a, reuse_b)
      c = __builtin_amdgcn_wmma_f32_16x16x4_f32(false, a, false, bb,
                                                (short)0, c, false, false);
    }

    #pragma unroll
    for (int j = 0; j < 8; ++j) {
      float lv = c[j] + pb;                            // logit (row 8*hi+j, col n)
      float gvj = (j < 4) ? g0[j] : g1[j - 4];
      float sv = lv + gvj;                             // logit + gumbel
      if (sv > stMax[j]) { stMax[j] = sv; stIdx[j] = n; }
      if (lv > dtMax[j]) { dtMax[j] = lv; dtIdx[j] = n; }
      // |logits| <= ~2 (w ~ N(0,0.02^2), K=256) -> unshifted exp is exact/safe
      float e = __expf(lv);
      S[j] += e;
      T[j] = fmaf(e, lv, T[j]);
    }
  }

  // Reduce across the 16 lanes of each half (xor masks < 16 stay in-half).
  #pragma unroll
  for (int j = 0; j < 8; ++j) {
    #pragma unroll
    for (int off = 8; off >= 1; off >>= 1) {
      S[j] += __shfl_xor(S[j], off, 32);
      T[j] += __shfl_xor(T[j], off, 32);
      float osv = __shfl_xor(stMax[j], off, 32);
      int   osi = __shfl_xor(stIdx[j], off, 32);
      if (osv > stMax[j] || (osv == stMax[j] && osi < stIdx[j])) { stMax[j] = osv; stIdx[j] = osi; }
      float odv = __shfl_xor(dtMax[j], off, 32);
      int   odi = __shfl_xor(dtIdx[j], off, 32);
      if (odv > dtMax[j] || (odv == dtMax[j] && odi < dtIdx[j])) { dtMax[j] = odv; dtIdx[j] = odi; }
    }
  }
  if (ln == 0) {
    #pragma unroll
    for (int j = 0; j < 8; ++j) {
      int m = 8 * hi + j;
      fstat[wv][m][0] = stMax[j]; fstat[wv][m][1] = dtMax[j];
      fstat[wv][m][2] = S[j];     fstat[wv][m][3] = T[j];
      istat[wv][m][0] = stIdx[j]; istat[wv][m][1] = dtIdx[j];
    }
  }
  __syncthreads();

  // Merge the 8 waves; min-index-on-tie reproduces jnp.argmax globally.
  if (t < 16) {
    const int m = t;
    float bsv = -3.4e38f, bdv = -3.4e38f, Ssum = 0.f, Tsum = 0.f;
    int bsi = 0x7fffffff, bdi = 0x7fffffff;
    for (int w = 0; w < 8; ++w) {
      float sv = fstat[w][m][0]; int si = istat[w][m][0];
      if (sv > bsv || (sv == bsv && si < bsi)) { bsv = sv; bsi = si; }
      float dv = fstat[w][m][1]; int di = istat[w][m][1];
      if (dv > bdv || (dv == bdv && di < bdi)) { bdv = dv; bdi = di; }
      Ssum += fstat[w][m][2]; Tsum += fstat[w][m][3];
    }
    const int pix = b * L + hw0 + m;
    const int sel = (mask[pix] > 0.5f) ? bsi : bdi;
    ind_ws[pix] = sel;
    ind_out[pix] = sel;
    // sum_n p*log(p*NE + 1e-10)  ->  T/S - log S + log NE  (eps negligible)
    float kl = Tsum / Ssum - __logf(Ssum) + __logf((float)NE);
    atomicAdd(accum, kl);
  }
}

// ---------------------------------------------------------------------------
// Kernel 3: z_q gather (codebook row per pixel) with LDS transpose so the
//           bdhw-layout stores are fully coalesced; also finalizes the loss.
// ---------------------------------------------------------------------------
__global__ __launch_bounds__(256) void gather_kernel(
    const float* __restrict__ embed_w, const int* __restrict__ ind,
    const float* __restrict__ accum, float* __restrict__ out) {
  __shared__ float tile[32 * 257];          // [pixel][d], pitch 257 (no conflicts)
  const int blk = blockIdx.x;               // 0..127 : 16 b x 8 strips of 32 hw
  const int b   = blk >> 3;
  const int hw0 = (blk & 7) * 32;
  const int t   = threadIdx.x;

  // Phase 1: coalesced 1KB codebook-row reads.
  for (int p = 0; p < 32; ++p) {
    int row = ind[b * L + hw0 + p];
    tile[p * 257 + t] = embed_w[(size_t)row * ED + t];
  }
  __syncthreads();

  // Phase 2: 128B-coalesced stores along hw (fastest dim of bdhw).
  const int hwo = t & 31;
  const int dg  = t >> 5;
  for (int db = 0; db < 32; ++db) {
    int d = dg * 32 + db;
    out[(size_t)b * ED * L + (size_t)d * L + hw0 + hwo] = tile[hwo * 257 + d];
  }
  if (blk == 0 && t == 0)
    out[(size_t)B_ * ED * L] = KLW * (*accum) / (float)(B_ * L);
}

// ---------------------------------------------------------------------------
extern "C" void kernel_launch(void* const* d_in, const int* in_sizes, int n_in,
                              void* d_out, int out_size, void* d_ws, size_t ws_size,
                              hipStream_t stream) {
  const float* z          = (const float*)d_in[0];
  const float* proj_w     = (const float*)d_in[1];
  const float* proj_b     = (const float*)d_in[2];
  const float* embed_w    = (const float*)d_in[3];
  const float* gumbel     = (const float*)d_in[4];
  const float* mask_noise = (const float*)d_in[5];

  float* outf   = (float*)d_out;
  // d_out layout: z_q [B*ED*L] floats, prior_loss [1], ind [B*L] int32 bits
  int*   ind_out = (int*)(outf + (size_t)B_ * ED * L + 1);

  float* maskp  = (float*)d_ws;
  int*   ind_ws = (int*)((float*)d_ws + 4096);
  float* accum  = (float*)d_ws + 8192;

  mask_kernel<<<B_, 256, 0, stream>>>(mask_noise, maskp, accum);
  quant_kernel<<<B_ * 16, 256, 0, stream>>>(z, proj_w, proj_b, gumbel, maskp,
                                            ind_ws, ind_out, accum);
  gather_kernel<<<B_ * 8, 256, 0, stream>>>(embed_w, ind_ws, accum, outf);
}